// PWGGNN_45174466019353
// MI455X (gfx1250) — compile-verified
//
#include <hip/hip_runtime.h>
#include <hip/hip_bf16.h>

typedef float v2f __attribute__((ext_vector_type(2)));
typedef float v4f __attribute__((ext_vector_type(4)));
typedef float v8f __attribute__((ext_vector_type(8)));

#define DIM 256
#define DIM2 512

// ---------------------------------------------------------------------------
// f32 atomic add helper (hardware global_atomic_add_f32 path)
// ---------------------------------------------------------------------------
__device__ __forceinline__ void atomAddF(float* p, float v) {
    unsafeAtomicAdd(p, v);
}

// ---------------------------------------------------------------------------
// Kernel 1: per-column sum / sum-of-squares over feat [N, 256]
// ---------------------------------------------------------------------------
__global__ void colstats_kernel(const float* __restrict__ feat,
                                float* __restrict__ sums, int N) {
    int col = threadIdx.x;
    float s = 0.f, ss = 0.f;
    for (int r = blockIdx.x; r < N; r += gridDim.x) {
        float v = feat[(size_t)r * DIM + col];
        s += v;
        ss += v * v;
    }
    atomAddF(&sums[col], s);
    atomAddF(&sums[DIM + col], ss);
}

// ---------------------------------------------------------------------------
// Kernel 2: BatchNorm (training mode, biased variance)
// ---------------------------------------------------------------------------
__global__ void bnorm_kernel(const float* __restrict__ feat,
                             const float* __restrict__ sums,
                             const float* __restrict__ gamma,
                             const float* __restrict__ beta,
                             float* __restrict__ h, int N) {
    int col = threadIdx.x;
    float invN  = 1.0f / (float)N;
    float mean  = sums[col] * invN;
    float var   = sums[DIM + col] * invN - mean * mean;
    float scale = rsqrtf(var + 1e-5f) * gamma[col];
    float shift = beta[col];
    for (int r = blockIdx.x; r < N; r += gridDim.x) {
        size_t idx = (size_t)r * DIM + col;
        h[idx] = (feat[idx] - mean) * scale + shift;
    }
}

// ---------------------------------------------------------------------------
// Kernel 3: WMMA fp32 GEMM with LDS-staged weight panel.
//   C[M,Nout](ldc) = A[M,K](lda) @ W[K,Nout](ldb) (+bias) (+=C if ACCUM)
//
// Block = 256 threads = 8 waves; each wave owns one 16-row M tile and a
// 128-column strip (8 x 16x16 accumulator tiles).  grid.y = Nout/128.
//
// W is staged chunk-by-chunk (KC=64 K-rows) into LDS in *fragment order*:
// for K-step ki, lane l, tile t, pair j the value
//     W[k0 + 4*ki + 2*(l>>4) + j][nbase + 16*t + (l&15)]
// lives at smem[(ki*32 + l)*20 + t*2 + j]   (20-float lane stride = padding).
// Inner loop: 1 global_load_b64 (A frag) + 4 ds_load_b128 (B frags) + 8 WMMA.
//
// ISA 7.12.2 fragment layouts (wave32):
//   A 16x4 f32 : lane = (k>=2?16:0)+m , vgpr = k%2
//   B 4x16 f32 : lane = (k>=2?16:0)+n , vgpr = k%2
//   C/D 16x16  : n = lane&15 , m = vgpr + (lane>=16 ? 8 : 0)
// ---------------------------------------------------------------------------
#define KC 64                       // K-rows per LDS chunk
#define LANE_STRIDE 20              // 16 data floats + 4 pad per (ki,lane)
#define SMEM_FLOATS ((KC / 4) * 32 * LANE_STRIDE)   // 10240 floats = 40 KB

template <bool ADD_BIAS, bool ACCUM>
__global__ void gemm_wmma_kernel(const float* __restrict__ A, int lda, int K,
                                 const float* __restrict__ W, int ldb,
                                 const float* __restrict__ bias,
                                 float* __restrict__ C, int ldc,
                                 int Mtiles) {
    __shared__ __align__(16) float smem[SMEM_FLOATS];

    const int tid   = threadIdx.x;
    const int lane  = tid & 31;
    const int wave  = tid >> 5;
    const int mtile = blockIdx.x * 8 + wave;          // 16-row tile index
    const bool active = (mtile < Mtiles);             // wave-uniform

    const int nhalf = lane >> 4;                      // K sub-pair select
    const int l15   = lane & 15;                      // m (A) / n (B,C)
    const int kofs  = nhalf * 2;
    const int nbase = blockIdx.y * 128;

    const float* __restrict__ Arow =
        A + (size_t)((active ? mtile : 0) * 16 + l15) * lda;

    v8f acc[8];
#pragma unroll
    for (int t = 0; t < 8; ++t) acc[t] = (v8f){};

    for (int k0 = 0; k0 < K; k0 += KC) {
        __syncthreads();   // previous chunk fully consumed
        // ---- cooperative fill: KC x 128 panel, coalesced reads ----
#pragma unroll 4
        for (int i = 0; i < (KC * 128) / 256; ++i) {
            int idx = tid + i * 256;
            int r = idx >> 7;          // 0..KC-1
            int c = idx & 127;         // 0..127
            float v = W[(size_t)(k0 + r) * ldb + nbase + c];
            int ki   = r >> 2;
            int rem  = r & 3;
            int dlane = (rem >> 1) * 16 + (c & 15);
            int j    = rem & 1;
            int t    = c >> 4;
            smem[(ki * 32 + dlane) * LANE_STRIDE + t * 2 + j] = v;
        }
        __syncthreads();

        if (active) {                  // wave-uniform: EXEC all-ones inside
#pragma unroll 4
            for (int ki = 0; ki < KC / 4; ++ki) {
                v2f af = *(const v2f*)(Arow + k0 + ki * 4 + kofs);

                const v4f* qb =
                    (const v4f*)smem + (size_t)(ki * 32 + lane) * (LANE_STRIDE / 4) * 0
                    + ((ki * 32 + lane) * LANE_STRIDE) / 4;
                v4f q0 = qb[0], q1 = qb[1], q2 = qb[2], q3 = qb[3];

                v2f b;
                b.x = q0[0]; b.y = q0[1];
                acc[0] = __builtin_amdgcn_wmma_f32_16x16x4_f32(false, af, false, b, (short)0, acc[0], false, false);
                b.x = q0[2]; b.y = q0[3];
                acc[1] = __builtin_amdgcn_wmma_f32_16x16x4_f32(false, af, false, b, (short)0, acc[1], false, false);
                b.x = q1[0]; b.y = q1[1];
                acc[2] = __builtin_amdgcn_wmma_f32_16x16x4_f32(false, af, false, b, (short)0, acc[2], false, false);
                b.x = q1[2]; b.y = q1[3];
                acc[3] = __builtin_amdgcn_wmma_f32_16x16x4_f32(false, af, false, b, (short)0, acc[3], false, false);
                b.x = q2[0]; b.y = q2[1];
                acc[4] = __builtin_amdgcn_wmma_f32_16x16x4_f32(false, af, false, b, (short)0, acc[4], false, false);
                b.x = q2[2]; b.y = q2[3];
                acc[5] = __builtin_amdgcn_wmma_f32_16x16x4_f32(false, af, false, b, (short)0, acc[5], false, false);
                b.x = q3[0]; b.y = q3[1];
                acc[6] = __builtin_amdgcn_wmma_f32_16x16x4_f32(false, af, false, b, (short)0, acc[6], false, false);
                b.x = q3[2]; b.y = q3[3];
                acc[7] = __builtin_amdgcn_wmma_f32_16x16x4_f32(false, af, false, b, (short)0, acc[7], false, false);
            }
        }
    }

    if (active) {
        // C/D layout: vgpr v -> row mtile*16 + v + (lane>=16 ? 8 : 0)
        const int mrow = mtile * 16 + (nhalf ? 8 : 0);
#pragma unroll
        for (int t = 0; t < 8; ++t) {
            const int n = nbase + t * 16 + l15;
            float bv = 0.f;
            if (ADD_BIAS) bv = bias[n];
            float* Cp = C + (size_t)mrow * ldc + n;
            v8f a = acc[t];
#pragma unroll
            for (int v = 0; v < 8; ++v) {
                float val = a[v] + bv;
                if (ACCUM) val += Cp[(size_t)v * ldc];
                Cp[(size_t)v * ldc] = val;
            }
        }
    }
}

// ---------------------------------------------------------------------------
// Kernel 4: edge gather-scale-scatter (both directions in one pass)
//   a[dst][0:256]   += F[src][0:256]   * iw   (a_in)
//   a[src][256:512] += F[dst][256:512] * ow   (a_out)
// ---------------------------------------------------------------------------
__global__ void scatter_kernel(const float* __restrict__ F,
                               const int* __restrict__ src,
                               const int* __restrict__ dst,
                               const float* __restrict__ iw,
                               const float* __restrict__ ow,
                               float* __restrict__ a) {
    const int e = blockIdx.x;
    const int t = threadIdx.x;
    const int s = src[e];
    const int d = dst[e];
    const float wi = iw[e];
    const float wo = ow[e];

    float fin  = F[(size_t)s * DIM2 + t] * wi;
    float fout = F[(size_t)d * DIM2 + DIM + t] * wo;
    atomAddF(&a[(size_t)d * DIM2 + t], fin);
    atomAddF(&a[(size_t)s * DIM2 + DIM + t], fout);
}

// ---------------------------------------------------------------------------
// Kernel 5: fused GRU-style update
//   G[:,0:256] = i_i+h_i ; G[:,256:512] = i_n+h_n
//   out = tanh(gn) + sigmoid(gi) * (h - tanh(gn))
// ---------------------------------------------------------------------------
__global__ void gru_kernel(const float* __restrict__ G,
                           const float* __restrict__ h,
                           float* __restrict__ out, int N) {
    size_t i = (size_t)blockIdx.x * blockDim.x + threadIdx.x;
    size_t total = (size_t)N * DIM;
    if (i >= total) return;
    size_t r = i >> 8;
    int    c = (int)(i & 255);
    float gi = G[r * DIM2 + c];
    float gn = G[r * DIM2 + DIM + c];
    float input_gate = 1.0f / (1.0f + __expf(-gi));
    float new_gate   = tanhf(gn);
    float hv = h[i];
    out[i] = new_gate + input_gate * (hv - new_gate);
}

// ---------------------------------------------------------------------------
// Host launch
// ---------------------------------------------------------------------------
extern "C" void kernel_launch(void* const* d_in, const int* in_sizes, int n_in,
                              void* d_out, int out_size, void* d_ws, size_t ws_size,
                              hipStream_t stream) {
    const float* feat  = (const float*)d_in[0];
    const int*   src   = (const int*)  d_in[1];
    const int*   dst   = (const int*)  d_in[2];
    const float* iw    = (const float*)d_in[3];
    const float* ow    = (const float*)d_in[4];
    const float* gamma = (const float*)d_in[5];
    const float* beta  = (const float*)d_in[6];
    const float* W_in  = (const float*)d_in[7];
    const float* b_in  = (const float*)d_in[8];
    const float* W_out = (const float*)d_in[9];
    const float* b_out = (const float*)d_in[10];
    const float* W_x2i = (const float*)d_in[11];
    const float* b_x2i = (const float*)d_in[12];
    const float* W_h2h = (const float*)d_in[13];

    const int N = in_sizes[0] / DIM;     // 100000
    const int E = in_sizes[1];           // 800000
    const int Mtiles = N / 16;           // 6250
    const int mblocks = (Mtiles + 7) / 8;

    // Workspace: h [N,256] | F/G [N,512] | a [N,512] | sums [512]
    float* h    = (float*)d_ws;
    float* F    = h + (size_t)N * DIM;
    float* a    = F + (size_t)N * DIM2;
    float* sums = a + (size_t)N * DIM2;
    float* G    = F;

    // 1) BatchNorm
    hipMemsetAsync(sums, 0, DIM2 * sizeof(float), stream);
    colstats_kernel<<<256, DIM, 0, stream>>>(feat, sums, N);
    bnorm_kernel<<<1024, DIM, 0, stream>>>(feat, sums, gamma, beta, h, N);

    // 2) F = [h@W_in+b_in | h@W_out+b_out]
    gemm_wmma_kernel<true, false><<<dim3(mblocks, DIM / 128), 256, 0, stream>>>(
        h, DIM, DIM, W_in, DIM, b_in, F, DIM2, Mtiles);
    gemm_wmma_kernel<true, false><<<dim3(mblocks, DIM / 128), 256, 0, stream>>>(
        h, DIM, DIM, W_out, DIM, b_out, F + DIM, DIM2, Mtiles);

    // 3) edge message passing: a = [a_in | a_out]
    hipMemsetAsync(a, 0, (size_t)N * DIM2 * sizeof(float), stream);
    scatter_kernel<<<E, DIM, 0, stream>>>(F, src, dst, iw, ow, a);

    // 4) G = a@W_x2i + b_x2i ; G += h@W_h2h   (G aliases F, which is dead)
    gemm_wmma_kernel<true, false><<<dim3(mblocks, DIM2 / 128), 256, 0, stream>>>(
        a, DIM2, DIM2, W_x2i, DIM2, b_x2i, G, DIM2, Mtiles);
    gemm_wmma_kernel<false, true><<<dim3(mblocks, DIM2 / 128), 256, 0, stream>>>(
        h, DIM, DIM, W_h2h, DIM2, nullptr, G, DIM2, Mtiles);

    // 5) GRU combine -> d_out
    const size_t total = (size_t)N * DIM;
    gru_kernel<<<(unsigned)((total + 255) / 256), 256, 0, stream>>>(G, h, (float*)d_out, N);
}